// FeatureField_64544768524376
// MI455X (gfx1250) — compile-verified
//
#include <hip/hip_runtime.h>
#include <hip/hip_bf16.h>
#include <stdint.h>

// Instant-NGP style multires hash encoding for MI455X (gfx1250, wave32).
//
// Roofline: ~144 MB HBM traffic (~6 us @ 23.3 TB/s); dominant cost is 128M
// data-dependent 8B gathers from a 4 MB table that is fully resident in the
// 192 MB L2 -> bound by VMEM issue rate / L2 gather bandwidth. No matmul
// structure -> WMMA inapplicable. Optimizations: one thread per (point,level)
// for perfectly coalesced float2 stores, single b96 coordinate load, unrolled
// 8x b64 gather clause, gfx1250 global_prefetch_b8 pipelining of the
// streaming coordinate reads across grid-stride iterations.

#define N_LEVELS 16
#define LOG2_T   19
#define T_MASK   ((1u << LOG2_T) - 1u)
#define PRIME1   2654435761u
#define PRIME2   805459861u

// floor(16 * 8^(l/15)) for l = 0..15, precomputed exactly (matches numpy fp64).
__device__ __constant__ float kRes[N_LEVELS] = {
    16.f, 18.f, 21.f, 24.f, 27.f, 32.f, 36.f, 42.f,
    48.f, 55.f, 64.f, 73.f, 84.f, 97.f, 111.f, 128.f
};

// 12-byte coordinate record: loads as a single global_load_b96 (4B aligned).
struct Coord3 { float x, y, z; };

__global__ __launch_bounds__(256) void FeatureField_64544768524376_kernel(
    const float* __restrict__ x,      // [N,3]
    const float* __restrict__ table,  // [2^19, 2]
    float*       __restrict__ out,    // [N, 32]
    int nPoints)
{
    const int total  = nPoints << 4;                    // N * 16 work items
    const int stride = (int)(blockDim.x * gridDim.x);   // multiple of 256
    const int idx0   = (int)(blockIdx.x * blockDim.x + threadIdx.x);
    if (idx0 >= total) return;

    // Level is loop-invariant per thread (stride % 16 == 0).
    const int   lvl = idx0 & (N_LEVELS - 1);
    const float res = kRes[lvl];

    const Coord3* __restrict__ xc   = (const Coord3*)x;
    const float2* __restrict__ tb   = (const float2*)table;
    float2*       __restrict__ out2 = (float2*)out;
    const int pstride = stride >> 4;

    for (int idx = idx0; idx < total; idx += stride) {
        const int p = idx >> 4;

        // gfx1250 path: prefetch next iteration's coordinates (global_prefetch_b8).
        const int pn = p + pstride;
        if (pn < nPoints) __builtin_prefetch((const float*)(xc + pn), 0, 3);

        // Single 12-byte (b96) coordinate load per item.
        const Coord3 c = xc[p];
        const float xs0 = c.x * res;
        const float xs1 = c.y * res;
        const float xs2 = c.z * res;

        const float fx = floorf(xs0), fy = floorf(xs1), fz = floorf(xs2);
        const float dx = xs0 - fx,    dy = xs1 - fy,    dz = xs2 - fz;

        // x in [0,1) -> coordinates nonnegative; uint32 wraparound mul matches
        // the reference's int32 overflow semantics bit-for-bit.
        const uint32_t ix = (uint32_t)(int)fx;
        const uint32_t iy = (uint32_t)(int)fy;
        const uint32_t iz = (uint32_t)(int)fz;

        // ceil replaced by floor+1: any corner whose index would differ has an
        // exactly-zero trilinear weight, so the blended sum is unchanged.
        const uint32_t hx0 = ix;
        const uint32_t hx1 = ix + 1u;
        const uint32_t hy0 = iy        * PRIME1;
        const uint32_t hy1 = (iy + 1u) * PRIME1;
        const uint32_t hz0 = iz        * PRIME2;
        const uint32_t hz1 = (iz + 1u) * PRIME2;

        // Corner order matches reference: 000,100,010,001,110,101,011,111.
        const uint32_t h0 = (hx0 ^ hy0 ^ hz0) & T_MASK;
        const uint32_t h1 = (hx1 ^ hy0 ^ hz0) & T_MASK;
        const uint32_t h2 = (hx0 ^ hy1 ^ hz0) & T_MASK;
        const uint32_t h3 = (hx0 ^ hy0 ^ hz1) & T_MASK;
        const uint32_t h4 = (hx1 ^ hy1 ^ hz0) & T_MASK;
        const uint32_t h5 = (hx1 ^ hy0 ^ hz1) & T_MASK;
        const uint32_t h6 = (hx0 ^ hy1 ^ hz1) & T_MASK;
        const uint32_t h7 = (hx1 ^ hy1 ^ hz1) & T_MASK;

        // Issue all 8 aligned b64 gathers back-to-back (L2-resident table).
        const float2 v0 = tb[h0];
        const float2 v1 = tb[h1];
        const float2 v2 = tb[h2];
        const float2 v3 = tb[h3];
        const float2 v4 = tb[h4];
        const float2 v5 = tb[h5];
        const float2 v6 = tb[h6];
        const float2 v7 = tb[h7];

        const float wx0 = 1.f - dx, wx1 = dx;
        const float wy0 = 1.f - dy, wy1 = dy;
        const float wz0 = 1.f - dz, wz1 = dz;

        const float w0 = wx0 * wy0 * wz0;
        const float w1 = wx1 * wy0 * wz0;
        const float w2 = wx0 * wy1 * wz0;
        const float w3 = wx0 * wy0 * wz1;
        const float w4 = wx1 * wy1 * wz0;
        const float w5 = wx1 * wy0 * wz1;
        const float w6 = wx0 * wy1 * wz1;
        const float w7 = wx1 * wy1 * wz1;

        float a0 = w0 * v0.x;
        float a1 = w0 * v0.y;
        a0 = fmaf(w1, v1.x, a0);  a1 = fmaf(w1, v1.y, a1);
        a0 = fmaf(w2, v2.x, a0);  a1 = fmaf(w2, v2.y, a1);
        a0 = fmaf(w3, v3.x, a0);  a1 = fmaf(w3, v3.y, a1);
        a0 = fmaf(w4, v4.x, a0);  a1 = fmaf(w4, v4.y, a1);
        a0 = fmaf(w5, v5.x, a0);  a1 = fmaf(w5, v5.y, a1);
        a0 = fmaf(w6, v6.x, a0);  a1 = fmaf(w6, v6.y, a1);
        a0 = fmaf(w7, v7.x, a0);  a1 = fmaf(w7, v7.y, a1);

        // out[p][lvl][0:2] — consecutive lanes write consecutive float2s:
        // a wave32 stores 256 B fully coalesced.
        out2[idx] = make_float2(a0, a1);
    }
}

extern "C" void kernel_launch(void* const* d_in, const int* in_sizes, int n_in,
                              void* d_out, int out_size, void* d_ws, size_t ws_size,
                              hipStream_t stream) {
    (void)n_in; (void)out_size; (void)d_ws; (void)ws_size;
    const float* x     = (const float*)d_in[0];   // [N,3] f32
    const float* table = (const float*)d_in[1];   // [2^19,2] f32
    float*       out   = (float*)d_out;           // [N,32] f32

    const int nPoints = in_sizes[0] / 3;
    const int total   = nPoints << 4;             // one thread per (point,level)

    const int threads = 256;                      // 8 wave32s per block
    int blocks = (total + threads - 1) / threads;
    if (blocks > 8192) blocks = 8192;             // grid-stride + prefetch pipelining
    if (blocks < 1) blocks = 1;

    hipLaunchKernelGGL(FeatureField_64544768524376_kernel,
                       dim3(blocks), dim3(threads), 0, stream,
                       x, table, out, nPoints);
}